// GraphAttentionLayer_88965952569378
// MI455X (gfx1250) — compile-verified
//
#include <hip/hip_runtime.h>
#include <hip/hip_bf16.h>
#include <stdint.h>

typedef __attribute__((ext_vector_type(2))) float v2f;
typedef __attribute__((ext_vector_type(8))) float v8f;

#define B_      8
#define N_      2048
#define FIN     128
#define FOUT    64
#define LRALPHA 0.2f
#define NEG_INF (-9.0e15f)
#define NCH     8              // row chunks for column-stats pass
#define CHROWS  (N_ / NCH)     // 256
#define JT      256            // columns per block in column-stats pass
#define JCH     64             // j chunk staged in LDS for pass C
#define NW32    (N_ / 32)      // 64 packed adjacency words per row

// ---------------------------------------------------------------------------
// Kernel 1: h = input @ W  (fp32 WMMA 16x16x4), plus f1 = h.a1, f2 = h.a2
// One wave per 16-row tile; 4 accumulator tiles cover Fout=64.
// ---------------------------------------------------------------------------
__global__ __launch_bounds__(256) void k_h_gemm(
    const float* __restrict__ inp, const float* __restrict__ W,
    const float* __restrict__ avec,
    float* __restrict__ h, float* __restrict__ f1, float* __restrict__ f2)
{
    __shared__ __align__(16) float Ws[FIN * FOUT];      // 32 KB
    const int t = threadIdx.x;

    // cooperative stage of W into LDS (8 float4 per thread)
    {
        const float4* Wg4 = (const float4*)W;
        float4* Ws4 = (float4*)Ws;
        #pragma unroll
        for (int q = 0; q < (FIN * FOUT / 4) / 256; ++q)
            Ws4[q * 256 + t] = Wg4[q * 256 + t];
    }
    __syncthreads();

    const int wave = t >> 5;
    const int lane = t & 31;
    const int m    = lane & 15;     // A-matrix row within tile
    const int half = lane >> 4;     // K sub-offset select
    const int rowbase = blockIdx.x * 128 + wave * 16;

    const float* ip = inp + (size_t)(rowbase + m) * FIN + 2 * half;

    v8f acc[4] = {};
    #pragma unroll 4
    for (int s = 0; s < FIN / 4; ++s) {
        const int k0 = 4 * s;
        v2f A;                       // A 16x4 f32: VGPR0 = K(2*half), VGPR1 = K(2*half)+1
        A.x = ip[k0];
        A.y = ip[k0 + 1];
        const float* wr0 = &Ws[(k0 + 2 * half) * FOUT + m];
        const float* wr1 = wr0 + FOUT;
        #pragma unroll
        for (int c = 0; c < 4; ++c) {
            v2f Bv;
            Bv.x = wr0[16 * c];
            Bv.y = wr1[16 * c];
            acc[c] = __builtin_amdgcn_wmma_f32_16x16x4_f32(
                false, A, false, Bv, (short)0, acc[c], false, false);
        }
    }

    // a1/a2 per lane column
    float a1v[4], a2v[4];
    #pragma unroll
    for (int c = 0; c < 4; ++c) {
        a1v[c] = avec[m + 16 * c];
        a2v[c] = avec[FOUT + m + 16 * c];
    }

    // C/D layout: VGPR r -> row r (lanes 0-15) / row r+8 (lanes 16-31)
    #pragma unroll
    for (int r = 0; r < 8; ++r) {
        const int row = rowbase + r + 8 * half;
        float p1 = 0.f, p2 = 0.f;
        #pragma unroll
        for (int c = 0; c < 4; ++c) {
            const float v = acc[c][r];
            h[(size_t)row * FOUT + m + 16 * c] = v;
            p1 += v * a1v[c];
            p2 += v * a2v[c];
        }
        #pragma unroll
        for (int d = 8; d >= 1; d >>= 1) {
            p1 += __shfl_xor(p1, d);
            p2 += __shfl_xor(p2, d);
        }
        if (m == 0) { f1[row] = p1; f2[row] = p2; }
    }
}

// ---------------------------------------------------------------------------
// Kernel 2: stream adj ONCE with non-temporal loads (128 MiB > L2; streamed,
// never re-read -> keep it out of L2 so h/packed stay resident). Per column j:
// online (max,sum) over a 256-row chunk; pack adjacency into bitmask words via
// wave32 ballot so pass C never re-reads adj.
// ---------------------------------------------------------------------------
__global__ __launch_bounds__(256) void k_colstats(
    const int* __restrict__ adj, const float* __restrict__ f1,
    const float* __restrict__ f2,
    uint32_t* __restrict__ packed, float* __restrict__ pmax,
    float* __restrict__ psum)
{
    __shared__ float f1s[CHROWS];
    const int t  = threadIdx.x;
    const int b  = blockIdx.z;
    const int ch = blockIdx.y;
    const int j  = blockIdx.x * JT + t;
    const int i0 = ch * CHROWS;

    f1s[t] = f1[(size_t)b * N_ + i0 + t];
    __syncthreads();

    const float f2j = f2[(size_t)b * N_ + j];
    const int wid  = t >> 5;
    const int lane = t & 31;
    const size_t adjbase = ((size_t)b * N_ + i0) * N_ + j;
    const size_t pkbase  = ((size_t)b * N_ + i0) * NW32 + (size_t)blockIdx.x * (JT / 32) + wid;

    float rm = -3.0e38f, rs = 0.f;
    for (int ii = 0; ii < CHROWS; ++ii) {
        const int av = __builtin_nontemporal_load(&adj[adjbase + (size_t)ii * N_]);
        const uint32_t mask = (uint32_t)__ballot(av > 0);
        if (lane == 0) packed[pkbase + (size_t)ii * NW32] = mask;

        float ev = f1s[ii] + f2j;
        ev = ev > 0.f ? ev : LRALPHA * ev;
        const float e  = (av > 0) ? ev : NEG_INF;
        const float nm = fmaxf(rm, e);
        rs = rs * __expf(rm - nm) + __expf(e - nm);
        rm = nm;
    }
    const size_t po = ((size_t)b * NCH + ch) * N_ + j;
    pmax[po] = rm;
    psum[po] = rs;
}

// ---------------------------------------------------------------------------
// Kernel 2b: reduce the NCH chunk partials -> colmax[b,j], 1/colsum[b,j]
// ---------------------------------------------------------------------------
__global__ __launch_bounds__(256) void k_colreduce(
    const float* __restrict__ pmax, const float* __restrict__ psum,
    float* __restrict__ colmax, float* __restrict__ colrcp)
{
    const int idx = blockIdx.x * 256 + threadIdx.x;   // == b*N + j
    const int b = idx >> 11;
    const int j = idx & (N_ - 1);
    float gm = -3.0e38f;
    #pragma unroll
    for (int c = 0; c < NCH; ++c)
        gm = fmaxf(gm, pmax[((size_t)b * NCH + c) * N_ + j]);
    float gs = 0.f;
    #pragma unroll
    for (int c = 0; c < NCH; ++c) {
        const size_t o = ((size_t)b * NCH + c) * N_ + j;
        gs += psum[o] * __expf(pmax[o] - gm);
    }
    colmax[idx] = gm;
    colrcp[idx] = 1.0f / gs;   // gs >= 1 always (max element contributes exp(0))
}

// ---------------------------------------------------------------------------
// Kernel 3: h' = attn @ h via fp32 WMMA. h chunks are staged into LDS with
// double-buffered GLOBAL_LOAD_ASYNC_TO_LDS_B128 (ASYNCcnt-tracked, bypasses
// VGPRs) so staging overlaps the WMMA loop. Attention weights rebuilt on the
// fly from f1/f2 + bitmask + column stats. Finish with ELU (NT stores).
// ---------------------------------------------------------------------------
__global__ __launch_bounds__(128) void k_attn_gemm(
    const float* __restrict__ h, const float* __restrict__ f1,
    const float* __restrict__ f2, const float* __restrict__ colmax,
    const float* __restrict__ colrcp, const uint32_t* __restrict__ packed,
    float* __restrict__ out)
{
    __shared__ __align__(16) float hs[2][JCH * FOUT];   // 2 x 16 KB
    __shared__ float f2s[2][JCH], cms[2][JCH], crs[2][JCH];

    const int t    = threadIdx.x;
    const int wave = t >> 5;
    const int lane = t & 31;
    const int m    = lane & 15;
    const int half = lane >> 4;
    const int b    = blockIdx.y;
    const int ibase = blockIdx.x * 64 + wave * 16;
    const int i     = ibase + m;                 // A-matrix row for this lane
    const size_t bN = (size_t)b * N_;

    const float f1i = f1[bN + i];
    const uint32_t* prow = packed + (bN + i) * NW32;

    v8f acc[4] = {};

    // Issue async staging of one 64-row h chunk into buffer `buf`.
    auto issue = [&](int jb, int buf) {
        const uint64_t gbase = (uint64_t)(uintptr_t)(h + (bN + jb) * FOUT);
        const uint32_t lbase = (uint32_t)(uintptr_t)&hs[buf][0];
        #pragma unroll
        for (int q = 0; q < (JCH * FOUT * 4) / (128 * 16); ++q) {   // 8 rounds
            const uint64_t ga = gbase + (uint64_t)((q * 128 + t) * 16);
            const uint32_t la = lbase + (uint32_t)((q * 128 + t) * 16);
            asm volatile("global_load_async_to_lds_b128 %0, %1, off"
                         :: "v"(la), "v"(ga) : "memory");
        }
        if (t < JCH) {
            f2s[buf][t] = f2[bN + jb + t];
            cms[buf][t] = colmax[bN + jb + t];
            crs[buf][t] = colrcp[bN + jb + t];
        }
    };

    issue(0, 0);   // prologue: fill buffer 0

    for (int jb = 0; jb < N_; jb += JCH) {
        const int cur = (jb / JCH) & 1;

        // wait for this wave's async loads, then make chunk visible block-wide
        asm volatile("s_wait_asynccnt 0" ::: "memory");
        __syncthreads();

        // overlap: stage the next chunk into the other buffer
        if (jb + JCH < N_) issue(jb + JCH, cur ^ 1);

        const uint32_t w0 = prow[(jb >> 5) + 0];
        const uint32_t w1 = prow[(jb >> 5) + 1];

        #pragma unroll 4
        for (int js = 0; js < JCH / 4; ++js) {
            const int jl0 = 4 * js + 2 * half;          // local j of A.x
            const int jl1 = jl0 + 1;
            const uint32_t bit0 = ((jl0 < 32 ? w0 : w1) >> (jl0 & 31)) & 1u;
            const uint32_t bit1 = ((jl1 < 32 ? w0 : w1) >> (jl1 & 31)) & 1u;

            float e0 = f1i + f2s[cur][jl0]; e0 = e0 > 0.f ? e0 : LRALPHA * e0;
            float e1 = f1i + f2s[cur][jl1]; e1 = e1 > 0.f ? e1 : LRALPHA * e1;
            const float m0 = bit0 ? e0 : NEG_INF;
            const float m1 = bit1 ? e1 : NEG_INF;

            v2f A;
            A.x = __expf(m0 - cms[cur][jl0]) * crs[cur][jl0];
            A.y = __expf(m1 - cms[cur][jl1]) * crs[cur][jl1];

            const float* hr0 = &hs[cur][(4 * js + 2 * half) * FOUT + m];
            const float* hr1 = hr0 + FOUT;
            #pragma unroll
            for (int c = 0; c < 4; ++c) {
                v2f Bv;
                Bv.x = hr0[16 * c];
                Bv.y = hr1[16 * c];
                acc[c] = __builtin_amdgcn_wmma_f32_16x16x4_f32(
                    false, A, false, Bv, (short)0, acc[c], false, false);
            }
        }
        // no trailing barrier needed: next iteration's wait+barrier orders
        // this iteration's reads of hs[cur] before any write into hs[cur]
    }

    // ELU + store (C/D layout: row r for lanes 0-15, r+8 for lanes 16-31)
    #pragma unroll
    for (int r = 0; r < 8; ++r) {
        const int row = ibase + r + 8 * half;
        #pragma unroll
        for (int c = 0; c < 4; ++c) {
            float v = acc[c][r];
            v = v > 0.f ? v : (__expf(v) - 1.f);
            __builtin_nontemporal_store(v, &out[(bN + row) * FOUT + m + 16 * c]);
        }
    }
}

// ---------------------------------------------------------------------------
extern "C" void kernel_launch(void* const* d_in, const int* in_sizes, int n_in,
                              void* d_out, int out_size, void* d_ws, size_t ws_size,
                              hipStream_t stream)
{
    const float* inp = (const float*)d_in[0];   // (8, 2048, 128) f32
    const int*   adj = (const int*)  d_in[1];   // (8, 2048, 2048) i32
    const float* W   = (const float*)d_in[2];   // (128, 64) f32
    const float* av  = (const float*)d_in[3];   // (128, 1) f32
    float* out = (float*)d_out;                 // (8, 2048, 64) f32

    // workspace carve-out (~9.7 MiB total)
    float* ws = (float*)d_ws;
    float* h    = ws; ws += (size_t)B_ * N_ * FOUT;      // 4 MiB
    float* f1   = ws; ws += (size_t)B_ * N_;
    float* f2   = ws; ws += (size_t)B_ * N_;
    float* cmax = ws; ws += (size_t)B_ * N_;
    float* crcp = ws; ws += (size_t)B_ * N_;
    float* pmax = ws; ws += (size_t)B_ * NCH * N_;
    float* psum = ws; ws += (size_t)B_ * NCH * N_;
    uint32_t* packed = (uint32_t*)ws;                    // 4 MiB bitmask

    k_h_gemm   <<<dim3((B_ * N_) / 128),      256, 0, stream>>>(inp, W, av, h, f1, f2);
    k_colstats <<<dim3(N_ / JT, NCH, B_),     256, 0, stream>>>(adj, f1, f2, packed, pmax, psum);
    k_colreduce<<<dim3((B_ * N_) / 256),      256, 0, stream>>>(pmax, psum, cmax, crcp);
    k_attn_gemm<<<dim3(N_ / 64, B_),          128, 0, stream>>>(h, f1, f2, cmax, crcp, packed, out);
}